// StreamingCVE_8289286881372
// MI455X (gfx1250) — compile-verified
//
#include <hip/hip_runtime.h>
#include <math.h>

// ---------------------------------------------------------------------------
// StreamingCVE correlation:  out = -|Σ(t-μ)s| / (sqrt(Σ(t-μ)²+ε)·sqrt(Σs²+ε))
// over θ = concat(p1,p2,p3) (fp32, n = 83.9M) and s = fp16 secret tiled to n.
// Single pass, 5 running sums, bandwidth-bound (≈14.4 µs at 23.3 TB/s).
// ---------------------------------------------------------------------------

typedef float    v2f __attribute__((ext_vector_type(2)));
typedef float    v8f __attribute__((ext_vector_type(8)));
typedef float    f4  __attribute__((ext_vector_type(4)));
typedef _Float16 h4  __attribute__((ext_vector_type(4)));

#define NBLK 1024
#define NTHR 256
#define WAVES_PER_BLK (NTHR / 32)
// Each wave processes 512 elements per chunk: 4 x (b128 theta) + 4 x (b64 s16)
#define CHUNK 512u

// Exact 32-lane wave sum using the f32 WMMA ones-matrix row-sum trick.
// A(16x4): vgpr0 = acc (lanes 0..31 cover A[i][0] and A[i][2]), vgpr1 = 0.
// B(4x16) = all ones  =>  D[i][j] = acc[i] + acc[i+16]  for every j.
// Lane j (j<16) holds D[0..7][j] in its 8 D registers; lane j+16 holds
// D[8..15][j]. Sum the 8 registers, then add across the 16-lane halves.
__device__ __forceinline__ float wave_sum_wmma(float v) {
  v2f a; a[0] = v;    a[1] = 0.0f;
  v2f b; b[0] = 1.0f; b[1] = 1.0f;
  v8f c = {};
  c = __builtin_amdgcn_wmma_f32_16x16x4_f32(false, a, false, b,
                                            (short)0, c, false, false);
  float r = ((c[0] + c[1]) + (c[2] + c[3])) + ((c[4] + c[5]) + (c[6] + c[7]));
  r += __shfl_xor(r, 16, 32);
  return r;                    // all lanes now hold the full 32-lane sum
}

struct Acc { float s1, s2, sts, ss, ss2; };

// Stream one parameter array (stream offset `base` into the tiled secret).
__device__ __forceinline__ void accum_stream(const float* __restrict__ t,
                                             unsigned len, unsigned base,
                                             const _Float16* __restrict__ sec,
                                             unsigned m, Acc& A,
                                             unsigned waveId, unsigned nWaves,
                                             unsigned lane, unsigned gtid,
                                             unsigned gthreads) {
  const unsigned lane4   = lane * 4u;
  const unsigned strideE = nWaves * CHUNK;        // elements per grid step
  const unsigned smod    = strideE % m;
  unsigned e = waveId * CHUNK;
  unsigned sBase = (base + e) % m;                // scalar (wave-uniform)

  for (; e + CHUNK <= len; e += strideE) {
    #pragma unroll
    for (int k = 0; k < 4; ++k) {
      // theta: wave-contiguous 512B per b128 instruction, NT (read-once)
      f4 tv = __builtin_nontemporal_load(
                  (const f4*)(t + e + (unsigned)k * 128u + lane4));
      // secret: group of 4 never straddles the wrap (m % 4 == 0)
      unsigned sk = sBase + (unsigned)k * 128u + lane4;
      sk = (sk >= m) ? (sk - m) : sk;
      h4 sv = *(const h4*)(sec + sk);
      #pragma unroll
      for (int i = 0; i < 4; ++i) {
        float x = tv[i];
        float y = (float)sv[i];
        A.s1  += x;
        A.s2   = fmaf(x, x, A.s2);
        A.sts  = fmaf(x, y, A.sts);
        A.ss  += y;
        A.ss2  = fmaf(y, y, A.ss2);
      }
    }
    sBase += smod;
    if (sBase >= m) sBase -= m;
  }

  // Generic scalar tail (empty for these sizes: len % 512 == 0)
  unsigned full = len - (len % CHUNK);
  for (unsigned i = full + gtid; i < len; i += gthreads) {
    float x = t[i];
    float y = (float)sec[(base + i) % m];
    A.s1 += x; A.s2 = fmaf(x, x, A.s2); A.sts = fmaf(x, y, A.sts);
    A.ss += y; A.ss2 = fmaf(y, y, A.ss2);
  }
}

__global__ __launch_bounds__(NTHR)
void corr_stream_kernel(const float* __restrict__ p1,
                        const float* __restrict__ p2,
                        const float* __restrict__ p3,
                        const _Float16* __restrict__ sec,
                        unsigned n1, unsigned n2, unsigned n3, unsigned m,
                        float* __restrict__ ws) {
  const unsigned tid      = threadIdx.x;
  const unsigned lane     = tid & 31u;
  const unsigned wave     = tid >> 5;
  const unsigned gtid     = blockIdx.x * NTHR + tid;
  const unsigned gthreads = NBLK * NTHR;
  const unsigned waveId   = gtid >> 5;
  const unsigned nWaves   = gthreads >> 5;

  Acc A = {0.f, 0.f, 0.f, 0.f, 0.f};
  accum_stream(p1, n1, 0u,       sec, m, A, waveId, nWaves, lane, gtid, gthreads);
  accum_stream(p2, n2, n1,       sec, m, A, waveId, nWaves, lane, gtid, gthreads);
  accum_stream(p3, n3, n1 + n2,  sec, m, A, waveId, nWaves, lane, gtid, gthreads);

  // Wave-level reduction on the matrix pipe (exact f32, EXEC all-ones here).
  float w1  = wave_sum_wmma(A.s1);
  float w2  = wave_sum_wmma(A.s2);
  float wts = wave_sum_wmma(A.sts);
  float wss = wave_sum_wmma(A.ss);
  float ws2 = wave_sum_wmma(A.ss2);

  __shared__ float red[WAVES_PER_BLK][5];
  if (lane == 0) {
    red[wave][0] = w1;  red[wave][1] = w2;  red[wave][2] = wts;
    red[wave][3] = wss; red[wave][4] = ws2;
  }
  __syncthreads();
  if (tid < 5) {
    float a = 0.f;
    #pragma unroll
    for (int w = 0; w < WAVES_PER_BLK; ++w) a += red[w][tid];
    ws[(size_t)tid * NBLK + blockIdx.x] = a;   // block partial, sum q = tid
  }
}

__global__ __launch_bounds__(NTHR)
void corr_finalize_kernel(const float* __restrict__ ws,
                          unsigned long long n, float* __restrict__ out) {
  __shared__ double sd[NTHR];
  double r[5];
  for (int q = 0; q < 5; ++q) {
    double a = 0.0;
    for (int i = threadIdx.x; i < NBLK; i += NTHR)
      a += (double)ws[(size_t)q * NBLK + i];
    sd[threadIdx.x] = a;
    __syncthreads();
    for (int off = NTHR / 2; off > 0; off >>= 1) {
      if ((int)threadIdx.x < off) sd[threadIdx.x] += sd[threadIdx.x + off];
      __syncthreads();
    }
    r[q] = sd[0];
    __syncthreads();
  }
  if (threadIdx.x == 0) {
    const double EPS = 1e-12;
    double S1 = r[0], S2 = r[1], Sts = r[2], Ss = r[3], Ss2 = r[4];
    double dn   = (double)n;
    double mean = S1 / dn;
    double num  = Sts - mean * Ss;            // Σ(t-μ)s
    double sth2 = S2 - S1 * S1 / dn;          // Σ(t-μ)²
    double corr = fabs(num) / (sqrt(sth2 + EPS) * sqrt(Ss2 + EPS));
    out[0] = (float)(-corr);
  }
}

extern "C" void kernel_launch(void* const* d_in, const int* in_sizes, int n_in,
                              void* d_out, int out_size, void* d_ws, size_t ws_size,
                              hipStream_t stream) {
  const float*    p1  = (const float*)d_in[0];
  const float*    p2  = (const float*)d_in[1];
  const float*    p3  = (const float*)d_in[2];
  const _Float16* sec = (const _Float16*)d_in[3];
  unsigned n1 = (unsigned)in_sizes[0];
  unsigned n2 = (unsigned)in_sizes[1];
  unsigned n3 = (unsigned)in_sizes[2];
  unsigned m  = (unsigned)in_sizes[3];
  float* ws = (float*)d_ws;                     // 5 * NBLK floats = 20 KB

  corr_stream_kernel<<<NBLK, NTHR, 0, stream>>>(p1, p2, p3, sec,
                                                n1, n2, n3, m, ws);
  unsigned long long n = (unsigned long long)n1 + n2 + n3;
  corr_finalize_kernel<<<1, NTHR, 0, stream>>>(ws, n, (float*)d_out);
}